// GCN_19997367730406
// MI455X (gfx1250) — compile-verified
//
#include <hip/hip_runtime.h>

typedef float v2f __attribute__((ext_vector_type(2)));
typedef float v8f __attribute__((ext_vector_type(8)));

#define NN 100000
#define EE 1000000
#define HH 64
#define GG 512
#define CC 2
#define SELF_W 2.0f

static inline int cdiv(int a, int b) { return (a + b - 1) / b; }

// ---------------- degree / norm precompute (layer-invariant) ----------------

__global__ void k_init_deg(float* __restrict__ deg) {
    int i = blockIdx.x * 256 + threadIdx.x;
    if (i < NN) deg[i] = SELF_W;
}

__global__ void k_deg_scatter(const int* __restrict__ dst,
                              const float* __restrict__ w,
                              float* __restrict__ deg) {
    int e = blockIdx.x * 256 + threadIdx.x;
    if (e < EE) unsafeAtomicAdd(&deg[dst[e]], w[e]);
}

__global__ void k_dinv(float* __restrict__ deg) {
    int i = blockIdx.x * 256 + threadIdx.x;
    if (i < NN) {
        float d = deg[i];
        deg[i] = (d > 0.0f) ? rsqrtf(d) : 0.0f;   // deg >= 2 always, guard anyway
    }
}

__global__ void k_norm(const int* __restrict__ src, const int* __restrict__ dst,
                       const float* __restrict__ w, const float* __restrict__ dinv,
                       float* __restrict__ norm) {
    int e = blockIdx.x * 256 + threadIdx.x;
    if (e < EE) norm[e] = dinv[src[e]] * w[e] * dinv[dst[e]];
}

// ---------------- WMMA f32 GEMM: out[n x 64] = act(A[n x 64]) @ W[64 x 64] ----------------
// Per wave: 16 rows x 64 cols via 4 accumulator tiles, K-loop of 16 steps of
// V_WMMA_F32_16X16X4_F32. Branch-free: A rows clamped (invalid rows never stored).
// W staged in LDS k-pair-interleaved so each B fragment is one ds_load_b64.
// Optional ReLU fused on the A load.

template <bool RELU>
__global__ __launch_bounds__(256) void k_gemm64(const float* __restrict__ A,
                                                const float* __restrict__ W,
                                                float* __restrict__ out, int n) {
    // LDS layout: wl[(k>>1)*128 + 2*c + (k&1)] = W[k][c]
    __shared__ float wl[64 * 64];
    const int tid  = threadIdx.x;
    const int wave = tid >> 5;            // 8 waves per block
    const int lane = tid & 31;
    const int m    = lane & 15;           // M index (A/D) / N column (B/D)
    const int hi   = lane >> 4;           // K-half select
    const int row_base = blockIdx.x * 128 + wave * 16;
    const int row  = row_base + m;
    const int rowc = (row < n) ? row : (n - 1);   // clamp: invalid rows never stored
    const float* arow = A + (size_t)rowc * HH;

    // Prefetch this lane's 16 A fragments (branch-free, overlaps LDS fill).
    v2f af[16];
#pragma unroll
    for (int s = 0; s < 16; ++s) af[s] = *(const v2f*)(arow + 4 * s + 2 * hi);
    if (RELU) {
#pragma unroll
        for (int s = 0; s < 16; ++s) {
            af[s][0] = fmaxf(af[s][0], 0.0f);
            af[s][1] = fmaxf(af[s][1], 0.0f);
        }
    }

    // Fill LDS (interleaved layout), float4 global reads.
#pragma unroll
    for (int j = 0; j < 4; ++j) {
        int idx = tid + 256 * j;          // float4 index 0..1023
        int k   = idx >> 4;               // 16 float4 per 64-col row
        int c4  = (idx & 15) * 4;
        float4 v = ((const float4*)W)[idx];
        float* p = wl + (k >> 1) * 128 + (k & 1) + 2 * c4;
        p[0] = v.x; p[2] = v.y; p[4] = v.z; p[6] = v.w;
    }
    __syncthreads();

    v8f c0 = {}, c1 = {}, c2 = {}, c3 = {};
#pragma unroll
    for (int s = 0; s < 16; ++s) {
        // k-step k0=4s; this lane's k-pair p = 2s+hi; col = 16t + m
        const float* bp = wl + (2 * s + hi) * 128 + 2 * m;
        v2f b0 = *(const v2f*)(bp + 0);
        v2f b1 = *(const v2f*)(bp + 32);
        v2f b2 = *(const v2f*)(bp + 64);
        v2f b3 = *(const v2f*)(bp + 96);
        c0 = __builtin_amdgcn_wmma_f32_16x16x4_f32(false, af[s], false, b0, (short)0, c0, false, false);
        c1 = __builtin_amdgcn_wmma_f32_16x16x4_f32(false, af[s], false, b1, (short)0, c1, false, false);
        c2 = __builtin_amdgcn_wmma_f32_16x16x4_f32(false, af[s], false, b2, (short)0, c2, false, false);
        c3 = __builtin_amdgcn_wmma_f32_16x16x4_f32(false, af[s], false, b3, (short)0, c3, false, false);
    }

    // D layout: VGPR r -> row M=r (lanes 0-15) / M=r+8 (lanes 16-31), col = lane&15
#pragma unroll
    for (int r = 0; r < 8; ++r) {
        const int orow = row_base + r + 8 * hi;
        if (orow < n) {
            float* op = out + (size_t)orow * HH + m;
            op[0]  = c0[r];
            op[16] = c1[r];
            op[32] = c2[r];
            op[48] = c3[r];
        }
    }
}

// ---------------- agg init: agg = (2*dinv^2)*xw + bias (self-loop + bias fused) ----------------

__global__ void k_init_agg(const float* __restrict__ xw, const float* __restrict__ dinv,
                           const float* __restrict__ bias, float* __restrict__ agg) {
    int t = blockIdx.x * 256 + threadIdx.x;   // NN*16 threads, float4 each
    int i = t >> 4;
    if (i >= NN) return;
    int f4 = t & 15;
    float di = dinv[i];
    float sl = SELF_W * di * di;
    float4 xv = ((const float4*)(xw + (size_t)i * HH))[f4];
    float4 bv = ((const float4*)bias)[f4];
    float4 r;
    r.x = sl * xv.x + bv.x;
    r.y = sl * xv.y + bv.y;
    r.z = sl * xv.z + bv.z;
    r.w = sl * xv.w + bv.w;
    ((float4*)(agg + (size_t)i * HH))[f4] = r;
}

// ---------------- edge scatter: agg[dst] += norm[e] * xw[src] ----------------
// One wave per edge; edge id forced into SGPR so src/dst/norm become scalar loads.
// float2 per lane -> coalesced 64-bit gathers, native f32 atomics.

__global__ __launch_bounds__(256) void k_edge_scatter(const int* __restrict__ src,
                                                      const int* __restrict__ dst,
                                                      const float* __restrict__ norm,
                                                      const float* __restrict__ xw,
                                                      float* __restrict__ agg) {
    int wlocal = __builtin_amdgcn_readfirstlane(threadIdx.x >> 5);
    int e = blockIdx.x * 8 + wlocal;          // 8 waves per block, 1 edge per wave
    if (e >= EE) return;
    int lane = threadIdx.x & 31;
    int f = lane * 2;
    int s = src[e], d = dst[e];
    float nv = norm[e];
    float2 v = *(const float2*)(xw + (size_t)s * HH + f);
    unsafeAtomicAdd(&agg[(size_t)d * HH + f + 0], nv * v.x);
    unsafeAtomicAdd(&agg[(size_t)d * HH + f + 1], nv * v.y);
}

// ---------------- pooling + classifier ----------------

__global__ void k_zero_pool(float* __restrict__ pooled, float* __restrict__ cnt) {
    int t = blockIdx.x * 256 + threadIdx.x;
    if (t < GG * HH) pooled[t] = 0.0f;
    if (t < GG) cnt[t] = 0.0f;
}

__global__ void k_count(const int* __restrict__ batch, float* __restrict__ cnt) {
    int i = blockIdx.x * 256 + threadIdx.x;
    if (i < NN) unsafeAtomicAdd(&cnt[batch[i]], 1.0f);
}

__global__ __launch_bounds__(256) void k_pool_scatter(const int* __restrict__ batch,
                                                      const float* __restrict__ h,
                                                      float* __restrict__ pooled) {
    int wlocal = __builtin_amdgcn_readfirstlane(threadIdx.x >> 5);
    int i = blockIdx.x * 8 + wlocal;          // 1 node per wave
    if (i >= NN) return;
    int lane = threadIdx.x & 31;
    int f = lane * 2;
    int g = batch[i];
    float2 v = *(const float2*)(h + (size_t)i * HH + f);
    unsafeAtomicAdd(&pooled[(size_t)g * HH + f + 0], v.x);
    unsafeAtomicAdd(&pooled[(size_t)g * HH + f + 1], v.y);
}

__global__ void k_final(const float* __restrict__ pooled, const float* __restrict__ cnt,
                        const float* __restrict__ Wl, const float* __restrict__ bl,
                        float* __restrict__ out) {
    int t = blockIdx.x * 256 + threadIdx.x;   // GG*CC threads
    if (t >= GG * CC) return;
    int g = t / CC, c = t % CC;
    float inv = 1.0f / fmaxf(cnt[g], 1.0f);
    float s = 0.0f;
    const float* pg = pooled + (size_t)g * HH;
#pragma unroll
    for (int f = 0; f < HH; ++f) s += pg[f] * Wl[f * CC + c];
    out[t] = s * inv + bl[c];
}

// ---------------- host launch ----------------

extern "C" void kernel_launch(void* const* d_in, const int* in_sizes, int n_in,
                              void* d_out, int out_size, void* d_ws, size_t ws_size,
                              hipStream_t stream) {
    const float* x    = (const float*)d_in[0];
    const int*   ei   = (const int*)d_in[1];
    const float* ew   = (const float*)d_in[2];
    const int*   bat  = (const int*)d_in[3];
    const float* W1   = (const float*)d_in[4];
    const float* b1   = (const float*)d_in[5];
    const float* W2   = (const float*)d_in[6];
    const float* b2   = (const float*)d_in[7];
    const float* W3   = (const float*)d_in[8];
    const float* b3   = (const float*)d_in[9];
    const float* Wl   = (const float*)d_in[10];
    const float* bl   = (const float*)d_in[11];
    float* out = (float*)d_out;

    const int* esrc = ei;        // edge_index[0, :]
    const int* edst = ei + EE;   // edge_index[1, :]

    // Workspace carve-up (float units, 64-float aligned)
    float* ws = (float*)d_ws;
    size_t off = 0;
    float* dinv   = ws + off; off += ((NN + 63) / 64) * 64;        // deg -> dinv in place
    float* normv  = ws + off; off += (size_t)EE;
    float* XW     = ws + off; off += (size_t)NN * HH;
    float* hA     = ws + off; off += (size_t)NN * HH;
    float* hB     = ws + off; off += (size_t)NN * HH;
    float* pooled = ws + off; off += (size_t)GG * HH;
    float* cnt    = ws + off; off += (size_t)GG;
    (void)ws_size; (void)n_in; (void)in_sizes; (void)out_size;

    const int TB = 256;
    const int gN    = cdiv(NN, TB);
    const int gE    = cdiv(EE, TB);
    const int gN16  = cdiv(NN * 16, TB);
    const int gE8   = cdiv(EE, 8);       // 1 wave per edge, 8 waves/block
    const int gN8   = cdiv(NN, 8);       // 1 wave per node
    const int gGemm = cdiv(NN, 128);

    // Degree / norm precompute (shared by all three layers)
    k_init_deg<<<gN, TB, 0, stream>>>(dinv);
    k_deg_scatter<<<gE, TB, 0, stream>>>(edst, ew, dinv);
    k_dinv<<<gN, TB, 0, stream>>>(dinv);
    k_norm<<<gE, TB, 0, stream>>>(esrc, edst, ew, dinv, normv);

    // Layer 1: x -> hA (no input ReLU)
    k_gemm64<false><<<gGemm, TB, 0, stream>>>(x, W1, XW, NN);
    k_init_agg<<<gN16, TB, 0, stream>>>(XW, dinv, b1, hA);
    k_edge_scatter<<<gE8, TB, 0, stream>>>(esrc, edst, normv, XW, hA);

    // Layer 2: relu(hA) -> hB
    k_gemm64<true><<<gGemm, TB, 0, stream>>>(hA, W2, XW, NN);
    k_init_agg<<<gN16, TB, 0, stream>>>(XW, dinv, b2, hB);
    k_edge_scatter<<<gE8, TB, 0, stream>>>(esrc, edst, normv, XW, hB);

    // Layer 3: relu(hB) -> hA
    k_gemm64<true><<<gGemm, TB, 0, stream>>>(hB, W3, XW, NN);
    k_init_agg<<<gN16, TB, 0, stream>>>(XW, dinv, b3, hA);
    k_edge_scatter<<<gE8, TB, 0, stream>>>(esrc, edst, normv, XW, hA);

    // Mean pool + classifier
    k_zero_pool<<<cdiv(GG * HH, TB), TB, 0, stream>>>(pooled, cnt);
    k_count<<<gN, TB, 0, stream>>>(bat, cnt);
    k_pool_scatter<<<gN8, TB, 0, stream>>>(bat, hA, pooled);
    k_final<<<cdiv(GG * CC, TB), TB, 0, stream>>>(pooled, cnt, Wl, bl, out);
}